// DiffeomorphicNetwork_24970939859046
// MI455X (gfx1250) — compile-verified
//
#include <hip/hip_runtime.h>
#include <stdint.h>

// Problem constants (B,D,H,W) = (2,128,128,128)
#define BATCH 2
#define DIM   128
#define HW    (128*128)          // 16384
#define DHW   (128*128*128)      // 2097152 = 1<<21
#define TW    8                  // output tile edge
#define HALO  2                  // |flow| <= 1.5 => corners within +/-2
#define TEXT  12                 // TW + 2*HALO
#define TVOL  (TEXT*TEXT*TEXT)   // 1728

typedef __attribute__((ext_vector_type(4))) unsigned int v4u;
typedef __attribute__((ext_vector_type(8))) int          v8i;
typedef __attribute__((ext_vector_type(4))) int          v4i;

// ---------------------------------------------------------------------------
// TDM: DMA a 12x12x12 fp32 sub-volume of a 128^3 tensor into contiguous LDS.
// Descriptor layout per CDNA5 ISA ch.8 (group0..3). Loop order: X innermost,
// so LDS receives tile[z][y][x] with x contiguous.
// ---------------------------------------------------------------------------
#if __has_builtin(__builtin_amdgcn_tensor_load_to_lds)
#define HAVE_TDM 1
static __device__ __forceinline__ void tdm_load_tile3d(uint32_t lds_addr, uint64_t gaddr) {
    v4u g0;
    g0[0] = 1u;                                   // count=1 (valid), no gather
    g0[1] = lds_addr;                             // LDS byte address
    g0[2] = (uint32_t)gaddr;                      // global_addr[31:0]
    g0[3] = (uint32_t)((gaddr >> 32) & 0x1FFFFFFu) | (2u << 30); // addr[56:32] | type=2

    v8i g1;
    g1[0] = (int)(2u << 16);                      // data_size = 4B; no pad/iterate/mcast
    g1[1] = (int)(128u << 16);                    // tensor_dim0 = 128 (bits 63:48)
    g1[2] = (int)(128u << 16);                    // tensor_dim0_hi=0 | tensor_dim1=128
    g1[3] = (int)((uint32_t)TEXT << 16);          // tensor_dim1_hi=0 | tile_dim0=12
    g1[4] = (int)((uint32_t)TEXT | ((uint32_t)TEXT << 16)); // tile_dim1, tile_dim2
    g1[5] = (int)128u;                            // tensor_dim0_stride = 128 (y stride)
    g1[6] = (int)((uint32_t)(HW & 0xFFFF) << 16); // stride0_hi=0 | stride1 low16
    g1[7] = (int)((uint32_t)HW >> 16);            // stride1 bits[47:16]

    v4i g2;
    g2[0] = 128;                                  // tensor_dim2 = 128
    g2[1] = 1;                                    // tensor_dim3 = 1 (unused)
    g2[2] = (int)DHW;                             // tensor_dim2_stride low32
    g2[3] = 0;                                    // stride2_hi | tile_dim3 = 0

    v4i g3 = {0, 0, 0, 0};

#if defined(__clang_major__) && __clang_major__ >= 23
    v8i g4 = {0, 0, 0, 0, 0, 0, 0, 0};
    __builtin_amdgcn_tensor_load_to_lds(g0, g1, g2, g3, g4, 0);
#else
    __builtin_amdgcn_tensor_load_to_lds(g0, g1, g2, g3, 0);
#endif
}
#else
#define HAVE_TDM 0
#endif

static __device__ __forceinline__ int iclamp(int v, int lo, int hi) {
    return v < lo ? lo : (v > hi ? hi : v);
}

// ---------------------------------------------------------------------------
// Kernel 1: flow0 = tanh(raw[:, :3]) * (1.5 / 64)
// ---------------------------------------------------------------------------
__global__ void __launch_bounds__(256)
init_flow_kernel(const float* __restrict__ raw, float* __restrict__ flow) {
    int i  = blockIdx.x * 256 + threadIdx.x;      // [0, B*3*DHW)
    int cl = i >> 21;                             // b*3 + c
    int s  = i & (DHW - 1);
    int b  = cl / 3;
    int c  = cl - b * 3;
    float v = raw[(((size_t)(b * 4 + c)) << 21) + s];
    flow[i] = tanhf(v) * (1.5f / 64.0f);
}

// ---------------------------------------------------------------------------
// Kernel 2: one scaling-and-squaring step: dst = src + warp(src, src).
// Per 8^3 tile: TDM-load 3 channel halo tiles (12^3) into LDS, then each
// thread does its 24 corner reads from LDS (ds_load).
// ---------------------------------------------------------------------------
__global__ void __launch_bounds__(512)
square_step_kernel(const float* __restrict__ src, float* __restrict__ dst) {
    __shared__ float tile[3][TEXT][TEXT][TEXT];   // 20736 B

    const int b  = (int)blockIdx.z >> 4;
    const int tz = ((int)blockIdx.z & 15) * TW;
    const int ty = (int)blockIdx.y * TW;
    const int tx = (int)blockIdx.x * TW;
    const int oz = iclamp(tz - HALO, 0, DIM - TEXT);
    const int oy = iclamp(ty - HALO, 0, DIM - TEXT);
    const int ox = iclamp(tx - HALO, 0, DIM - TEXT);

#if HAVE_TDM
    {
        const uint32_t lds0 = (uint32_t)(uintptr_t)&tile[0][0][0][0];
        if (threadIdx.x == 0 && threadIdx.y == 0 && threadIdx.z == 0) {
            for (int c = 0; c < 3; ++c) {
                const float* g = src + (((size_t)(b * 3 + c)) << 21)
                                     + ((size_t)oz << 14) + (oy << 7) + ox;
                tdm_load_tile3d(lds0 + (uint32_t)c * (TVOL * 4), (uint64_t)(uintptr_t)g);
            }
        }
        __builtin_amdgcn_s_wait_tensorcnt(0);
        // TDM wrote LDS behind the compiler's back: declare memory modified so
        // the ds_loads below can't be folded/hoisted.
        asm volatile("" ::: "memory");
    }
#else
    {
        const int ftid = (int)threadIdx.x + (int)threadIdx.y * 8 + (int)threadIdx.z * 64;
        for (int e = ftid; e < 3 * TVOL; e += 512) {
            int c = e / TVOL, r = e - c * TVOL;
            int z = r / (TEXT * TEXT), y = (r / TEXT) % TEXT, x = r % TEXT;
            tile[c][z][y][x] = src[(((size_t)(b * 3 + c)) << 21)
                                   + ((size_t)(oz + z) << 14) + ((oy + y) << 7) + (ox + x)];
        }
    }
#endif
    __syncthreads();

    const int gz = tz + (int)threadIdx.z;
    const int gy = ty + (int)threadIdx.y;
    const int gx = tx + (int)threadIdx.x;
    const int cz = gz - oz, cy = gy - oy, cx = gx - ox;

    const float f0 = tile[0][cz][cy][cx];   // x displacement
    const float f1 = tile[1][cz][cy][cx];   // y displacement
    const float f2 = tile[2][cz][cy][cx];   // z displacement

    const float px = fminf(fmaxf((float)gx + f0, 0.0f), 127.0f);
    const float py = fminf(fmaxf((float)gy + f1, 0.0f), 127.0f);
    const float pz = fminf(fmaxf((float)gz + f2, 0.0f), 127.0f);

    const float fx0 = floorf(px), fy0 = floorf(py), fz0 = floorf(pz);
    const float wx = px - fx0, wy = py - fy0, wz = pz - fz0;
    int x0 = (int)fx0, y0 = (int)fy0, z0 = (int)fz0;
    int x1 = min(x0 + 1, 127), y1 = min(y0 + 1, 127), z1 = min(z0 + 1, 127);
    const int ax0 = x0 - ox, ax1 = x1 - ox;
    const int ay0 = y0 - oy, ay1 = y1 - oy;
    const int az0 = z0 - oz, az1 = z1 - oz;

    const size_t obase = (((size_t)b * 3) << 21) + ((size_t)gz << 14) + (gy << 7) + gx;
    const float ctr[3] = {f0, f1, f2};
#pragma unroll
    for (int c = 0; c < 3; ++c) {
        float c000 = tile[c][az0][ay0][ax0], c001 = tile[c][az0][ay0][ax1];
        float c010 = tile[c][az0][ay1][ax0], c011 = tile[c][az0][ay1][ax1];
        float c100 = tile[c][az1][ay0][ax0], c101 = tile[c][az1][ay0][ax1];
        float c110 = tile[c][az1][ay1][ax0], c111 = tile[c][az1][ay1][ax1];
        float a00 = fmaf(wx, c001 - c000, c000);
        float a01 = fmaf(wx, c011 - c010, c010);
        float a10 = fmaf(wx, c101 - c100, c100);
        float a11 = fmaf(wx, c111 - c110, c110);
        float b0  = fmaf(wy, a01 - a00, a00);
        float b1  = fmaf(wy, a11 - a10, a10);
        float s   = fmaf(wz, b1 - b0, b0);
        dst[obase + ((size_t)c << 21)] = ctr[c] + s;
    }
}

// ---------------------------------------------------------------------------
// Kernel 3: warp x[:,1] with the final flow, SDF logit, topology-corrected
// sigmoid. Same TDM halo-tiling (single channel).
// ---------------------------------------------------------------------------
__global__ void __launch_bounds__(512)
final_warp_kernel(const float* __restrict__ xin, const float* __restrict__ raw,
                  const float* __restrict__ flow, float* __restrict__ out) {
    __shared__ float xt[TEXT][TEXT][TEXT];        // 6912 B

    const int b  = (int)blockIdx.z >> 4;
    const int tz = ((int)blockIdx.z & 15) * TW;
    const int ty = (int)blockIdx.y * TW;
    const int tx = (int)blockIdx.x * TW;
    const int oz = iclamp(tz - HALO, 0, DIM - TEXT);
    const int oy = iclamp(ty - HALO, 0, DIM - TEXT);
    const int ox = iclamp(tx - HALO, 0, DIM - TEXT);

#if HAVE_TDM
    {
        const uint32_t lds0 = (uint32_t)(uintptr_t)&xt[0][0][0];
        if (threadIdx.x == 0 && threadIdx.y == 0 && threadIdx.z == 0) {
            const float* g = xin + (((size_t)(b * 2 + 1)) << 21)
                                 + ((size_t)oz << 14) + (oy << 7) + ox;
            tdm_load_tile3d(lds0, (uint64_t)(uintptr_t)g);
        }
        __builtin_amdgcn_s_wait_tensorcnt(0);
        asm volatile("" ::: "memory");
    }
#else
    {
        const int ftid = (int)threadIdx.x + (int)threadIdx.y * 8 + (int)threadIdx.z * 64;
        for (int e = ftid; e < TVOL; e += 512) {
            int z = e / (TEXT * TEXT), y = (e / TEXT) % TEXT, x = e % TEXT;
            xt[z][y][x] = xin[(((size_t)(b * 2 + 1)) << 21)
                              + ((size_t)(oz + z) << 14) + ((oy + y) << 7) + (ox + x)];
        }
    }
#endif
    __syncthreads();

    const int gz = tz + (int)threadIdx.z;
    const int gy = ty + (int)threadIdx.y;
    const int gx = tx + (int)threadIdx.x;
    const size_t idx = ((size_t)gz << 14) + (gy << 7) + gx;
    const size_t fb  = ((size_t)b * 3) << 21;

    const float f0 = flow[fb + idx];
    const float f1 = flow[fb + ((size_t)1 << 21) + idx];
    const float f2 = flow[fb + ((size_t)2 << 21) + idx];

    const float px = fminf(fmaxf((float)gx + f0, 0.0f), 127.0f);
    const float py = fminf(fmaxf((float)gy + f1, 0.0f), 127.0f);
    const float pz = fminf(fmaxf((float)gz + f2, 0.0f), 127.0f);

    const float fx0 = floorf(px), fy0 = floorf(py), fz0 = floorf(pz);
    const float wx = px - fx0, wy = py - fy0, wz = pz - fz0;
    int x0 = (int)fx0, y0 = (int)fy0, z0 = (int)fz0;
    int x1 = min(x0 + 1, 127), y1 = min(y0 + 1, 127), z1 = min(z0 + 1, 127);
    const int ax0 = x0 - ox, ax1 = x1 - ox;
    const int ay0 = y0 - oy, ay1 = y1 - oy;
    const int az0 = z0 - oz, az1 = z1 - oz;

    float c000 = xt[az0][ay0][ax0], c001 = xt[az0][ay0][ax1];
    float c010 = xt[az0][ay1][ax0], c011 = xt[az0][ay1][ax1];
    float c100 = xt[az1][ay0][ax0], c101 = xt[az1][ay0][ax1];
    float c110 = xt[az1][ay1][ax0], c111 = xt[az1][ay1][ax1];
    float a00 = fmaf(wx, c001 - c000, c000);
    float a01 = fmaf(wx, c011 - c010, c010);
    float a10 = fmaf(wx, c101 - c100, c100);
    float a11 = fmaf(wx, c111 - c110, c110);
    float b0  = fmaf(wy, a01 - a00, a00);
    float b1  = fmaf(wy, a11 - a10, a10);
    float w   = fmaf(wz, b1 - b0, b0);

    const float rt = raw[(((size_t)(b * 4 + 3)) << 21) + idx];
    const float sdf   = logf(w + 1e-4f) - logf(1.0f - w + 1e-4f);
    const float t     = 1.0f / (1.0f + expf(-rt));
    const float delta = tanhf(rt);                 // MAX_TOPO_OFFSET = 1.0
    out[((size_t)b << 21) + idx] = 1.0f / (1.0f + expf(-(sdf + delta * t)));
}

// ---------------------------------------------------------------------------
// Host launcher. Workspace: two ping-pong flow fields (B,3,D,H,W) fp32,
// 48 MB each (both resident in the 192 MB L2).
// ---------------------------------------------------------------------------
extern "C" void kernel_launch(void* const* d_in, const int* in_sizes, int n_in,
                              void* d_out, int out_size, void* d_ws, size_t ws_size,
                              hipStream_t stream) {
    (void)in_sizes; (void)n_in; (void)out_size; (void)ws_size;
    const float* x   = (const float*)d_in[0];   // (2,2,128,128,128)
    const float* raw = (const float*)d_in[1];   // (2,4,128,128,128)
    float* out = (float*)d_out;                 // (2,1,128,128,128)

    float* fA = (float*)d_ws;
    float* fB = fA + (size_t)BATCH * 3 * DHW;   // ping-pong halves of workspace

    const int n_flow = BATCH * 3 * DHW;         // 12,582,912 (multiple of 256)
    init_flow_kernel<<<n_flow / 256, 256, 0, stream>>>(raw, fA);

    dim3 blk(TW, TW, TW);                       // 512 threads = 16 wave32
    dim3 grd(DIM / TW, DIM / TW, (DIM / TW) * BATCH);

    const float* src = fA;
    float* dst = fB;
    for (int i = 0; i < 6; ++i) {               // N_STEPS = 6
        square_step_kernel<<<grd, blk, 0, stream>>>(src, dst);
        const float* t = dst; dst = (float*)src; src = t;
    }
    final_warp_kernel<<<grd, blk, 0, stream>>>(x, raw, src, out);
}